// TemporalPropagator_56650618634828
// MI455X (gfx1250) — compile-verified
//
#include <hip/hip_runtime.h>
#include <math.h>

// ---------------------------------------------------------------------------
// TemporalPropagator for MI455X (gfx1250, wave32, WMMA).
// Precision: fp32 end-to-end, matrix ops via V_WMMA_F32_16X16X4_F32.
// Workspace requirement: ~160 MB (weights + planar complex intermediates).
// ---------------------------------------------------------------------------

typedef float v2f __attribute__((ext_vector_type(2)));
typedef float v8f __attribute__((ext_vector_type(8)));

#define DIMN   256
#define BROWS  16384          // B*S = 8*2048 tokens

// ======================= setup: Cayley via Newton–Schulz ====================

__global__ void buildA(const float* __restrict__ raw_re, const float* __restrict__ raw_im,
                       float* __restrict__ Ar, float* __restrict__ Ai) {
  int i = blockIdx.x, j = threadIdx.x;
  Ar[i * DIMN + j] = 0.5f * (raw_re[i * DIMN + j] - raw_re[j * DIMN + i]);
  Ai[i * DIMN + j] = 0.5f * (raw_im[i * DIMN + j] + raw_im[j * DIMN + i]);
}

__global__ void initX(const float* __restrict__ Ar, const float* __restrict__ Ai,
                      float* __restrict__ Xr, float* __restrict__ Xi) {
  int i = blockIdx.x, j = threadIdx.x;
  Xr[i * DIMN + j] = Ar[i * DIMN + j] + (i == j ? 1.0f : 0.0f);
  Xi[i * DIMN + j] = Ai[i * DIMN + j];
}

// C = alpha*(A@B) + beta*D   (complex, planar, DIMN x DIMN)
__global__ void cgemm(const float* __restrict__ Ar, const float* __restrict__ Ai,
                      const float* __restrict__ Br, const float* __restrict__ Bi,
                      const float* __restrict__ Dr, const float* __restrict__ Di,
                      float* __restrict__ Cr, float* __restrict__ Ci,
                      float alpha, float beta) {
  __shared__ float Asr[16][17], Asi[16][17], Bsr[16][17], Bsi[16][17];
  int tx = threadIdx.x, ty = threadIdx.y;
  int row = blockIdx.y * 16 + ty, col = blockIdx.x * 16 + tx;
  float ar = 0.f, ai = 0.f;
  for (int kt = 0; kt < DIMN; kt += 16) {
    Asr[ty][tx] = Ar[row * DIMN + kt + tx];
    Asi[ty][tx] = Ai[row * DIMN + kt + tx];
    Bsr[ty][tx] = Br[(kt + ty) * DIMN + col];
    Bsi[ty][tx] = Bi[(kt + ty) * DIMN + col];
    __syncthreads();
#pragma unroll
    for (int k = 0; k < 16; ++k) {
      float xr = Asr[ty][k], xi = Asi[ty][k];
      float br = Bsr[k][tx], bi = Bsi[k][tx];
      ar += xr * br - xi * bi;
      ai += xr * bi + xi * br;
    }
    __syncthreads();
  }
  Cr[row * DIMN + col] = alpha * ar + beta * Dr[row * DIMN + col];
  Ci[row * DIMN + col] = alpha * ai + beta * Di[row * DIMN + col];
}

// Wt_enc[k][n], n<256: Re(M_inv)[n][k]; n>=256: Im(M_inv)[n-256][k]
// M_inv[e,d] = sum_j V[e,j] * exp(-ls[j]) * conj(U[d,j])
__global__ void assembleEnc(const float* __restrict__ Vr, const float* __restrict__ Vi,
                            const float* __restrict__ Ur, const float* __restrict__ Ui,
                            const float* __restrict__ ls, float* __restrict__ Wt) {
  int d = blockIdx.x, e = threadIdx.x;
  float ar = 0.f, ai = 0.f;
  for (int j = 0; j < DIMN; ++j) {
    float s  = expf(-ls[j]);
    float vr = Vr[e * DIMN + j] * s, vi = Vi[e * DIMN + j] * s;
    float ur = Ur[d * DIMN + j],     ui = Ui[d * DIMN + j];
    ar += vr * ur + vi * ui;
    ai += vi * ur - vr * ui;
  }
  Wt[d * 2 * DIMN + e]        = ar;
  Wt[d * 2 * DIMN + DIMN + e] = ai;
}

// Decode weight as real 2x2 block form: out = [re|im] of h_next @ M^T (complex)
// M[e,d] = sum_j U[e,j] * exp(ls[j]) * conj(V[d,j])
__global__ void assembleDec(const float* __restrict__ Ur, const float* __restrict__ Ui,
                            const float* __restrict__ Vr, const float* __restrict__ Vi,
                            const float* __restrict__ ls, float* __restrict__ Wt) {
  int d = blockIdx.x, e = threadIdx.x;
  float ar = 0.f, ai = 0.f;
  for (int j = 0; j < DIMN; ++j) {
    float s  = expf(ls[j]);
    float ur = Ur[e * DIMN + j] * s, ui = Ui[e * DIMN + j] * s;
    float vr = Vr[d * DIMN + j],     vi = Vi[d * DIMN + j];
    ar += ur * vr + ui * vi;
    ai += ui * vr - ur * vi;
  }
  const int N2 = 2 * DIMN;
  Wt[d * N2 + e]                   = ar;   // k=d (h_re)  -> out_re
  Wt[d * N2 + DIMN + e]            = ai;   // k=d (h_re)  -> out_im
  Wt[(d + DIMN) * N2 + e]          = -ai;  // k=d+256 (h_im) -> out_re
  Wt[(d + DIMN) * N2 + DIMN + e]   = ar;   // k=d+256 (h_im) -> out_im
}

__global__ void transposeGate(const float* __restrict__ W, float* __restrict__ Wt) {
  int i = blockIdx.x * blockDim.x + threadIdx.x;      // 65536
  int k = i >> 8, n = i & 255;
  Wt[k * DIMN + n] = W[n * DIMN + k];
}

__global__ void transposeLam(const float* __restrict__ W, float* __restrict__ Wt) {
  int i = blockIdx.x * blockDim.x + threadIdx.x;      // 131072, W is [512,256]
  int k = i >> 9, n = i & 511;
  Wt[k * 2 * DIMN + n] = W[n * DIMN + k];
}

// ========================= fp32 WMMA GEMM ===================================
// out[m, n] = sum_k X[m, k] * Wt[k, n],  N = NT*128, M-tile = MT*16 rows.
// Block = 256 threads (8 waves). MT 16-row M-subtiles staged in LDS (pad +4
// kills the 16-way bank conflict of 1KB rows). Each wave owns NT 16-wide
// n-tiles; B fragments come straight from global (weights <=1MB stay hot in
// the 192MB L2) and are reused across the MT subtiles, halving L2 traffic and
// load-issue rate vs MT=1. All tile loops are compile-time so accumulators
// are statically indexed VGPRs (no v_movrel) and fragment loops fully unroll.
template <int KDIM, int NT, int MT>
__global__ __launch_bounds__(256) void wmmaGemmF32(
    const float* __restrict__ X, const float* __restrict__ Wt, float* __restrict__ Out,
    int planeA, int rowStride, int colStride) {
  constexpr int N = NT * 128;
  constexpr int ldsStride = KDIM + 4;
  constexpr int k4 = KDIM >> 2;
  __shared__ float lds[MT * 16 * ldsStride];

  const int m0 = blockIdx.x * (MT * 16);
  const int tid = threadIdx.x;

  // stage X[MT*16][KDIM] (coalesced b128 loads)
  const float4* X4 = (const float4*)(X + (size_t)m0 * KDIM);
#pragma unroll
  for (int i = tid; i < MT * 16 * k4; i += 256) {
    int r = i / k4, c = i % k4;                    // k4 is a power of two
    float4 v = X4[r * k4 + c];
    *(float4*)&lds[r * ldsStride + 4 * c] = v;
  }
  __syncthreads();

  const int lane = tid & 31;
  const int wave = tid >> 5;
  const int hi   = lane >> 4;      // half-wave: K pair select / upper M half
  const int r    = lane & 15;
  const int n0   = wave * (NT * 16);

  v8f acc[MT][NT] = {};

  __builtin_prefetch(Wt, 0, 1);    // global_prefetch_b8, warm weight lines

  const float* wcol = Wt + (size_t)2 * hi * N;     // rows ks+2hi, ks+2hi+1
  const float* arow = &lds[r * ldsStride + 2 * hi];

#pragma unroll 2
  for (int ks = 0; ks < KDIM; ks += 4) {
    // A fragments (16x4 f32): lane L vgpr v holds X[r][ks + v + 2*hi]
    v2f a[MT];
#pragma unroll
    for (int s = 0; s < MT; ++s)
      a[s] = *(const v2f*)(arow + s * 16 * ldsStride + ks);
    const float* w0 = wcol + (size_t)ks * N;
    const float* w1 = w0 + N;
    v2f b[NT];
#pragma unroll
    for (int t = 0; t < NT; ++t) {
      int c = n0 + t * 16 + r;
      b[t].x = w0[c];              // B[k=ks+2hi  ][n]
      b[t].y = w1[c];              // B[k=ks+2hi+1][n]
    }
#pragma unroll
    for (int s = 0; s < MT; ++s)
#pragma unroll
      for (int t = 0; t < NT; ++t) {
        acc[s][t] = __builtin_amdgcn_wmma_f32_16x16x4_f32(
            false, a[s], false, b[t], (short)0, acc[s][t], false, false);
      }
  }

  // C/D layout: vgpr v -> row v + 8*hi, col = lane&15
#pragma unroll
  for (int s = 0; s < MT; ++s)
#pragma unroll
    for (int t = 0; t < NT; ++t) {
      int col = n0 + t * 16 + r;
      size_t base = (size_t)(col >> 8) * (size_t)planeA + (size_t)(col & 255) * colStride;
#pragma unroll
      for (int v = 0; v < 8; ++v) {
        int row = m0 + s * 16 + v + 8 * hi;
        Out[base + (size_t)row * rowStride] = acc[s][t][v];
      }
    }
}

// ========================= elementwise kernels ==============================

__global__ void gateApply(float* __restrict__ xg, const float* __restrict__ glin,
                          const float* __restrict__ gate_b, int plane) {
  int idx = blockIdx.x * blockDim.x + threadIdx.x;
  int d = idx & 255;
  float g = glin[idx] + gate_b[d];
  float s = g / (1.0f + expf(-g));          // silu
  xg[idx]         *= s;
  xg[idx + plane] *= s;
}

__global__ void stateUpdate(const float* __restrict__ ht, const float* __restrict__ xg,
                            const float* __restrict__ dyn, const float* __restrict__ dt,
                            const float* __restrict__ noise_re, const float* __restrict__ noise_im,
                            const float* __restrict__ ld, const float* __restrict__ lf,
                            const float* __restrict__ law_re, const float* __restrict__ law_im,
                            const float* __restrict__ src_re, const float* __restrict__ src_im,
                            const float* __restrict__ raw_noise, const float* __restrict__ lam_b,
                            const float* __restrict__ dt_scales, float* __restrict__ hnext,
                            int plane) {
  int idx = blockIdx.x * blockDim.x + threadIdx.x;
  int m = idx >> 8, d = idx & 255;

  float dte = dt[m] * dt_scales[d];                 // DT_REF = 1
  float lre = -expf(ld[d]) + law_re[d];
  float lim = lf[d] + law_im[d];
  float dynr = tanhf(dyn[idx] + lam_b[d]);
  float dyni = tanhf(dyn[idx + plane] + lam_b[d + 256]);
  float Zr = (lre + dynr) * dte;
  float Zi = (lim + dyni) * dte;

  float cy = cosf(Zi), sy = sinf(Zi);
  float er = expf(Zr);
  float az2 = Zr * Zr + Zi * Zi;
  float p1r, p1i;
  if (az2 < 1e-8f) {                                // |Z| < 1e-4
    p1r = 1.0f + 0.5f * Zr;
    p1i = 0.5f * Zi;
  } else {                                          // cexpm1(Z)/Z
    float sh  = sinf(0.5f * Zi);
    float cre = expm1f(Zr) * cy - 2.0f * sh * sh;
    float cim = er * sy;
    float inv = 1.0f / az2;
    p1r = (cre * Zr + cim * Zi) * inv;
    p1i = (cim * Zr - cre * Zi) * inv;
  }
  p1r *= dte; p1i *= dte;                           // op_phi1
  float decr = er * cy, deci = er * sy;             // op_decay = exp(Z)

  float xfr = xg[idx] + src_re[d];
  float xfi = xg[idx + plane] + src_im[d];
  float utr = xfr * p1r - xfi * p1i;
  float uti = xfr * p1i + xfi * p1r;

  float ns  = log1pf(expf(raw_noise[0])) + 1e-6f;   // softplus
  float den = 2.0f * lre;
  if (fabsf(den) < 1e-6f) den = 1e-6f;
  float var = ns * ns * expm1f(2.0f * lre * dte) / den;
  float sdv = sqrtf(fabsf(var));
  const float inv_s2 = 0.70710678118654752f;
  float cr = noise_re[idx] * inv_s2, ci = noise_im[idx] * inv_s2;

  float htr = ht[idx], hti = ht[idx + plane];
  float hnr = htr * decr - hti * deci + utr + cr * sdv;
  float hni = htr * deci + hti * decr + uti + ci * sdv;

  hnext[(size_t)m * 512 + d]       = hnr;           // K-contiguous [re|im] row
  hnext[(size_t)m * 512 + 256 + d] = hni;
}

// ============================== launch ======================================

extern "C" void kernel_launch(void* const* d_in, const int* in_sizes, int n_in,
                              void* d_out, int out_size, void* d_ws, size_t ws_size,
                              hipStream_t stream) {
  const float* h_prev    = (const float*)d_in[0];
  const float* x_input   = (const float*)d_in[1];
  const float* dt        = (const float*)d_in[2];
  const float* noise_re  = (const float*)d_in[3];
  const float* noise_im  = (const float*)d_in[4];
  const float* u_raw_re  = (const float*)d_in[5];
  const float* u_raw_im  = (const float*)d_in[6];
  const float* v_raw_re  = (const float*)d_in[7];
  const float* v_raw_im  = (const float*)d_in[8];
  const float* log_sigma = (const float*)d_in[9];
  const float* ld        = (const float*)d_in[10];
  const float* lf        = (const float*)d_in[11];
  const float* law_re    = (const float*)d_in[12];
  const float* law_im    = (const float*)d_in[13];
  const float* src_re    = (const float*)d_in[14];
  const float* src_im    = (const float*)d_in[15];
  const float* raw_noise = (const float*)d_in[16];
  const float* lam_W     = (const float*)d_in[17];
  const float* lam_b     = (const float*)d_in[18];
  const float* gate_W    = (const float*)d_in[19];
  const float* gate_b    = (const float*)d_in[20];
  const float* dt_scales = (const float*)d_in[21];
  float* out = (float*)d_out;

  float* ws = (float*)d_ws;
  size_t off = 0;
  auto alloc = [&](size_t n) { float* p = ws + off; off += n; return p; };
  const size_t MM = 65536;                  // 256*256 plane

  float* Wt_enc  = alloc(256 * 512);
  float* Wt_dec  = alloc(512 * 512);
  float* Wt_gate = alloc(256 * 256);
  float* Wt_lam  = alloc(256 * 512);
  float* Ar = alloc(MM); float* Ai = alloc(MM);
  float* Xr = alloc(MM); float* Xi = alloc(MM);
  float* Yr = alloc(MM); float* Yi = alloc(MM);
  float* Tr = alloc(MM); float* Ti = alloc(MM);
  float* Ur = alloc(MM); float* Ui = alloc(MM);
  float* Vr = alloc(MM); float* Vi = alloc(MM);

  const int plane = BROWS * 256;
  float* h_tilde = alloc((size_t)2 * plane);
  float* xg      = alloc((size_t)2 * plane);
  float* glin    = alloc((size_t)plane);
  float* dynb    = alloc((size_t)2 * plane);
  float* hnext   = alloc((size_t)BROWS * 512);

  dim3 g16(16, 16), b16(16, 16);

  // ---- Cayley U = solve(I-A, I+A) via Newton–Schulz (4 quadratic iters) ----
  const float* raws[2][2] = {{u_raw_re, u_raw_im}, {v_raw_re, v_raw_im}};
  float* dsts[2][2] = {{Ur, Ui}, {Vr, Vi}};
  for (int which = 0; which < 2; ++which) {
    buildA<<<256, 256, 0, stream>>>(raws[which][0], raws[which][1], Ar, Ai);
    initX<<<256, 256, 0, stream>>>(Ar, Ai, Xr, Xi);            // X0 = I + A
    float *xr = Xr, *xi = Xi, *yr = Yr, *yi = Yi;
    for (int it = 0; it < 4; ++it) {
      // T = (I - A) X = X - A*X
      cgemm<<<g16, b16, 0, stream>>>(Ar, Ai, xr, xi, xr, xi, Tr, Ti, -1.0f, 1.0f);
      // Xn = 2X - X*T
      cgemm<<<g16, b16, 0, stream>>>(xr, xi, Tr, Ti, xr, xi, yr, yi, -1.0f, 2.0f);
      float* t;
      t = xr; xr = yr; yr = t;
      t = xi; xi = yi; yi = t;
    }
    // U = X * (I + A) = X*A + X
    cgemm<<<g16, b16, 0, stream>>>(xr, xi, Ar, Ai, xr, xi,
                                   dsts[which][0], dsts[which][1], 1.0f, 1.0f);
  }

  assembleEnc<<<256, 256, 0, stream>>>(Vr, Vi, Ur, Ui, log_sigma, Wt_enc);
  assembleDec<<<256, 256, 0, stream>>>(Ur, Ui, Vr, Vi, log_sigma, Wt_dec);
  transposeGate<<<256, 256, 0, stream>>>(gate_W, Wt_gate);
  transposeLam<<<512, 256, 0, stream>>>(lam_W, Wt_lam);

  // ---- main pipeline over 16384 tokens ----
  const int nblk = BROWS / 32;              // MT=2 -> 32 rows per block

  // encode h: [16384,256] @ Wt_enc[256,512] -> planar complex
  wmmaGemmF32<256, 4, 2><<<nblk, 256, 0, stream>>>(h_prev, Wt_enc, h_tilde, plane, 256, 1);
  // encode x
  wmmaGemmF32<256, 4, 2><<<nblk, 256, 0, stream>>>(x_input, Wt_enc, xg, plane, 256, 1);
  // gate logits from Re(x_tilde)
  wmmaGemmF32<256, 2, 2><<<nblk, 256, 0, stream>>>(xg, Wt_gate, glin, 0, 256, 1);
  gateApply<<<BROWS, 256, 0, stream>>>(xg, glin, gate_b, plane);
  // dyn = gated Re(x_tilde) @ lam_W^T  -> planar [dyn_re | dyn_im]
  wmmaGemmF32<256, 4, 2><<<nblk, 256, 0, stream>>>(xg, Wt_lam, dynb, plane, 256, 1);
  stateUpdate<<<BROWS, 256, 0, stream>>>(h_tilde, xg, dynb, dt, noise_re, noise_im,
                                         ld, lf, law_re, law_im, src_re, src_im,
                                         raw_noise, lam_b, dt_scales, hnext, plane);
  // decode: [16384,512] @ Wt_dec[512,512] -> interleaved complex64 output
  wmmaGemmF32<512, 4, 2><<<nblk, 256, 0, stream>>>(hnext, Wt_dec, out, 1, 512, 2);
}